// Laplace_78056735638168
// MI455X (gfx1250) — compile-verified
//
#include <hip/hip_runtime.h>
#include <cmath>

// Problem constants from the reference.
#define T_DIM 256
#define B_DIM 4
#define D_DIM 64
#define G_DIM 4
#define NT    32   // N_TAUS

typedef unsigned int       u32;
typedef unsigned long long u64;
typedef u32 u32x4 __attribute__((ext_vector_type(4)));
typedef int i32x4 __attribute__((ext_vector_type(4)));
typedef int i32x8 __attribute__((ext_vector_type(8)));

#if defined(__gfx1250__)
// Issue one TDM 2D tile load: `rows` rows of `row_elems` fp32 each,
// global row stride `stride_elems`, packed contiguously into LDS at
// byte offset lds_off. Descriptor packed per CDNA5 ISA.
// This toolchain exposes the 6-arg builtin: (g0, g1, g2, g3, g4, cpol).
__device__ __forceinline__ void tdm_load_tile_f32(u32 lds_off, const void* gptr,
                                                  u32 row_elems, u32 rows,
                                                  u32 stride_elems)
{
    u64 ga = (u64)(uintptr_t)gptr;
    u32x4 g0;
    g0.x = 1u;                                             // count=1 valid user D#
    g0.y = lds_off;                                        // lds_addr (bytes)
    g0.z = (u32)ga;                                        // global_addr[31:0]
    g0.w = (u32)((ga >> 32) & 0x01FFFFFFu) | (2u << 30);   // global_addr[56:32] | type=2
    i32x8 g1;
    g1[0] = (int)(2u << 16);            // workgroup_mask=0 (not in cluster), data_size=2 (4B)
    g1[1] = (int)(row_elems << 16);     // atomic_barrier_addr=0 | tensor_dim0[15:0]
    g1[2] = (int)(rows << 16);          // tensor_dim0[31:16]=0 | tensor_dim1[15:0]
    g1[3] = (int)(row_elems << 16);     // tensor_dim1[31:16]=0 | tile_dim0
    g1[4] = (int)rows;                  // tile_dim1 | tile_dim2=0 (unused -> 2D)
    g1[5] = (int)stride_elems;          // tensor_dim0_stride[31:0]
    g1[6] = 0;                          // stride hi | tensor_dim1_stride lo
    g1[7] = 0;
    i32x4 gz4 = (i32x4){0, 0, 0, 0};            // groups 2/3: tensor_dim2..4 unused
    i32x8 gz8 = (i32x8){0, 0, 0, 0, 0, 0, 0, 0};// trailing group (6-arg form): zero
    __builtin_amdgcn_tensor_load_to_lds(g0, g1, gz4, gz4, gz8, 0);
}
#endif

__global__ __launch_bounds__(256) void laplace_scan_kernel(
    const float* __restrict__ f,
    const float* __restrict__ alpha,
    double* __restrict__ out)
{
    // Per-block staging: f/alpha for (all T) x (8 d-columns), compact layout [t*8 + dl].
    __shared__ float smemF[T_DIM * 8];
    __shared__ float smemA[T_DIM * 8];

    const int b  = blockIdx.x >> 3;         // batch
    const int d0 = (blockIdx.x & 7) << 3;   // 8 d-columns per block
    const int j  = threadIdx.x & 31;        // lane = tau index (wave32)
    const int dl = threadIdx.x >> 5;        // wave  = local d column
    const int d  = d0 + dl;

    const float* fsrc = f     + b * D_DIM + d0;
    const float* asrc = alpha + b * D_DIM + d0;

#if defined(__gfx1250__)
    // Wave 0 DMAs both tiles via the Tensor Data Mover, then drains TENSORcnt.
    if (threadIdx.x < 32) {
        tdm_load_tile_f32((u32)(uintptr_t)&smemF[0], fsrc, 8u, (u32)T_DIM,
                          (u32)(B_DIM * D_DIM));
        tdm_load_tile_f32((u32)(uintptr_t)&smemA[0], asrc, 8u, (u32)T_DIM,
                          (u32)(B_DIM * D_DIM));
        __builtin_amdgcn_s_wait_tensorcnt(0);
    }
#else
    // Portable fallback (host parse / non-gfx1250): plain staged copy.
    for (int i = threadIdx.x; i < T_DIM * 8; i += 256) {
        int t = i >> 3, dd = i & 7;
        smemF[i] = fsrc[t * (B_DIM * D_DIM) + dd];
        smemA[i] = asrc[t * (B_DIM * D_DIM) + dd];
    }
#endif
    __syncthreads();

    // tau_star = 100^(j/31)  =>  1/tau = exp(-j * ln(100)/31)
    const double inv_tau = exp(-(double)j * (4.605170185988091 / 31.0));

    // |x_m| = r*sqrt(1+m^2) for the exprel Taylor-switch (|x| > 0.002).
    const double absfac[G_DIM] = {1.0, 1.4142135623730951,
                                  2.23606797749979, 3.1622776601683795};

    double hr[G_DIM], hi[G_DIM];
#pragma unroll
    for (int m = 0; m < G_DIM; ++m) { hr[m] = 0.0; hi[m] = 0.0; }

    const int out_base = b * (D_DIM * NT) + d * NT + j;   // (b,d,j) within one t-slab

    for (int t = 0; t < T_DIM; ++t) {
        const double a  = (double)smemA[t * 8 + dl];
        const double fv = (double)smemF[t * 8 + dl];
        const double r  = a * inv_tau;                    // s*alpha = r*(1 + i m)

        double sn, cs;
#if defined(__HIP_DEVICE_COMPILE__)
        sincos(r, &sn, &cs);
#else
        sn = sin(r); cs = cos(r);
#endif
        const double e = exp(-r);

        // hh_m = exp(-r) * (cos r - i sin r)^m : 1 exp + 1 sincos covers all 4 poles.
        double hhr[G_DIM], hhi[G_DIM];
        hhr[0] = e;                        hhi[0] = 0.0;
        hhr[1] = e * cs;                   hhi[1] = -(e * sn);
        hhr[2] = hhr[1]*cs + hhi[1]*sn;    hhi[2] = hhi[1]*cs - hhr[1]*sn;
        hhr[3] = hhr[2]*cs + hhi[2]*sn;    hhi[3] = hhi[2]*cs - hhr[2]*sn;

        double acc = 0.0;
#pragma unroll
        for (int m = 0; m < G_DIM; ++m) {
            const double md   = (double)m;
            const double m2p1 = 1.0 + md * md;

            // exact: exprel(x) = (hh - 1)/x,  1/x = -(1 - i m)/(r*(1+m^2))
            const double wr = hhr[m] - 1.0;
            const double wi = hhi[m];
            const double inv_den = 1.0 / (r * m2p1);
            const double exr = -(wr + wi * md) * inv_den;
            const double exi =  (wr * md - wi) * inv_den;

            // Taylor fallback near 0: 1 + x/2 + x^2/6 + x^3/24 + x^4/120, x = -r - i m r
            const double xr = -r, xi = -md * r;
            double pr = 1.0 / 24.0 + xr * (1.0 / 120.0);
            double pi =              xi * (1.0 / 120.0);
            double qr = 1.0 / 6.0 + (xr * pr - xi * pi);
            double qi =             (xr * pi + xi * pr);
            pr = 0.5 + (xr * qr - xi * qi);
            pi =       (xr * qi + xi * qr);
            qr = 1.0 + (xr * pr - xi * pi);
            qi =       (xr * pi + xi * pr);

            const bool exact = (r * absfac[m]) > 0.002;
            const double br = fv * (exact ? exr : qr);
            const double bi = fv * (exact ? exi : qi);

            // h = h*hh + b
            const double nr = hr[m] * hhr[m] - hi[m] * hhi[m] + br;
            const double ni = hr[m] * hhi[m] + hi[m] * hhr[m] + bi;
            hr[m] = nr; hi[m] = ni;
            acc += nr;                         // sum of real parts over poles
        }

        // til_f_acc[t,b,d,j] = (2/G) * sum_m Re(h): 32 contiguous f64 per wave.
        out[(size_t)t * (B_DIM * D_DIM * NT) + out_base] = 0.5 * acc;
    }

    // Final carry hx[b,d,m,j] as complex128 (re,im interleaved), after til_f_acc.
    double* hx = out + (size_t)T_DIM * B_DIM * D_DIM * NT;
    const size_t hx_base = ((size_t)(b * D_DIM + d) * G_DIM) * NT + (size_t)j;
#pragma unroll
    for (int m = 0; m < G_DIM; ++m) {
        hx[2 * (hx_base + (size_t)m * NT) + 0] = hr[m];
        hx[2 * (hx_base + (size_t)m * NT) + 1] = hi[m];
    }
}

extern "C" void kernel_launch(void* const* d_in, const int* in_sizes, int n_in,
                              void* d_out, int out_size, void* d_ws, size_t ws_size,
                              hipStream_t stream)
{
    (void)in_sizes; (void)n_in; (void)d_ws; (void)ws_size; (void)out_size;
    const float* f     = (const float*)d_in[0];
    const float* alpha = (const float*)d_in[1];
    double* out        = (double*)d_out;

    // 32 blocks x 256 threads (8 wave32 each): block = (b, 8 d-columns),
    // wave = d-column, lane = tau. Covers B*D*NT = 8192 recurrences.
    laplace_scan_kernel<<<dim3(B_DIM * (D_DIM / 8)), dim3(256), 0, stream>>>(f, alpha, out);
}